// Upsample_27839978013207
// MI455X (gfx1250) — compile-verified
//
#include <hip/hip_runtime.h>

typedef __attribute__((ext_vector_type(2))) float v2f;
typedef __attribute__((ext_vector_type(8))) float v8f;

// ---------------------------------------------------------------------------
// Kernel 1: build out_coords = [coords ; coords+(sx,sy) ; coords+(0,sy) ; coords+(sx,0)]
// ---------------------------------------------------------------------------
__global__ __launch_bounds__(256) void prep_coords_kernel(
    const float* __restrict__ coords,
    const float* __restrict__ spacing,
    float* __restrict__ out_coords, int n)
{
    int i = blockIdx.x * blockDim.x + threadIdx.x;
    if (i >= n) return;
    float x = coords[2 * i + 0];
    float y = coords[2 * i + 1];
    float sx = spacing[0];
    float sy = spacing[1];
    out_coords[2 * i + 0] = x;
    out_coords[2 * i + 1] = y;
    float* p1 = out_coords + 2 * (size_t)n;   // set 1: (x+sx, y+sy)
    p1[2 * i + 0] = x + sx;
    p1[2 * i + 1] = y + sy;
    float* p2 = out_coords + 4 * (size_t)n;   // set 2: (x, y+sy)
    p2[2 * i + 0] = x;
    p2[2 * i + 1] = y + sy;
    float* p3 = out_coords + 6 * (size_t)n;   // set 3: (x+sx, y)
    p3[2 * i + 0] = x + sx;
    p3[2 * i + 1] = y;
}

// ---------------------------------------------------------------------------
// Kernel 2: copy original values block (vectorized float4)
// ---------------------------------------------------------------------------
__global__ __launch_bounds__(256) void copy_values_kernel(
    const float4* __restrict__ src, float4* __restrict__ dst, int n4)
{
    int i = blockIdx.x * blockDim.x + threadIdx.x;
    if (i < n4) dst[i] = src[i];
}

// ---------------------------------------------------------------------------
// Kernel 3: precompute B fragments in WMMA lane layout.
// Tile t, lanes 0-15 (K0,K1):  (-2*cx, -2*cy) of column t*16+lane
// Tile t, lanes 16-31 (K2,K3): (cx^2+cy^2, 0) of column t*16+(lane-16)
// ---------------------------------------------------------------------------
__global__ __launch_bounds__(256) void bfrag_prep_kernel(
    const float* __restrict__ coords, float2* __restrict__ bfrag, int n)
{
    int i = blockIdx.x * blockDim.x + threadIdx.x;
    if (i >= n) return;
    float cx = coords[2 * i + 0];
    float cy = coords[2 * i + 1];
    int tile = i >> 4, c = i & 15;
    bfrag[tile * 32 + c]      = make_float2(-2.0f * cx, -2.0f * cy);
    bfrag[tile * 32 + 16 + c] = make_float2(cx * cx + cy * cy, 0.0f);
}

// ---------------------------------------------------------------------------
// Kernel 4: partial argmin via WMMA. One wave = (16 query rows) x (one column
// chunk of tiles_per_chunk 16-wide tiles). 2 WMMAs in flight per iteration.
// Writes (bestd, bestn) per row per chunk to `partial`.
// ---------------------------------------------------------------------------
__global__ __launch_bounds__(256) void nn_partial_kernel(
    const v2f*   __restrict__ bfrag,     // [n_tiles][32]
    const float* __restrict__ qcoords,   // n_queries x 2
    const float* __restrict__ shift,     // 2
    float2*      __restrict__ partial,   // [n_queries][split]
    int tiles_per_chunk, int split)
{
    const int lane = threadIdx.x & 31;
    const int wave = threadIdx.x >> 5;
    const int col  = lane & 15;
    const int hi   = lane >> 4;

    const int task  = blockIdx.x * 8 + wave;
    const int chunk = task % split;
    const int m0    = (task / split) * 16;

    // A fragment: row m = [qx, qy, 1, 0]
    const int m  = m0 + col;
    const float qx = qcoords[2 * m + 0] - shift[0];
    const float qy = qcoords[2 * m + 1] - shift[1];
    v2f a;
    a.x = hi ? 1.0f : qx;
    a.y = hi ? 0.0f : qy;

    float bestd[8];
    int   bestn[8];
#pragma unroll
    for (int r = 0; r < 8; ++r) { bestd[r] = 3.402823466e38f; bestn[r] = 0; }

    const int tile0 = chunk * tiles_per_chunk;
    for (int tt = 0; tt < tiles_per_chunk; tt += 2) {     // tiles_per_chunk is even
        const int tile = tile0 + tt;
        const v2f b0 = bfrag[tile * 32 + lane];
        const v2f b1 = bfrag[(tile + 1) * 32 + lane];
        const v8f cz = {0.f, 0.f, 0.f, 0.f, 0.f, 0.f, 0.f, 0.f};
        const v8f d0 = __builtin_amdgcn_wmma_f32_16x16x4_f32(
            false, a, false, b0, (short)0, cz, false, false);
        const v8f d1 = __builtin_amdgcn_wmma_f32_16x16x4_f32(
            false, a, false, b1, (short)0, cz, false, false);
        const int n0 = tile * 16 + col;
#pragma unroll
        for (int r = 0; r < 8; ++r) {
            const bool lt1 = d1[r] < d0[r];               // strict: prefer lower n
            const float dm = lt1 ? d1[r] : d0[r];
            const int   nm = n0 + (lt1 ? 16 : 0);
            const bool lt = dm < bestd[r];
            bestn[r] = lt ? nm : bestn[r];
            bestd[r] = lt ? dm : bestd[r];
        }
    }

    // argmin across the 16 lanes of each wave-half (tie -> lower index)
#pragma unroll
    for (int s = 1; s <= 8; s <<= 1) {
#pragma unroll
        for (int r = 0; r < 8; ++r) {
            const float od = __shfl_xor(bestd[r], s, 32);
            const int   on = __shfl_xor(bestn[r], s, 32);
            const bool take = (od < bestd[r]) || (od == bestd[r] && on < bestn[r]);
            bestd[r] = take ? od : bestd[r];
            bestn[r] = take ? on : bestn[r];
        }
    }
    // half 0: rows m0+0..7 in slots 0..7; half 1: rows m0+8..15
    if (col == 0) {
#pragma unroll
        for (int r = 0; r < 8; ++r) {
            const int row = m0 + r + (hi ? 8 : 0);
            partial[(size_t)row * split + chunk] =
                make_float2(bestd[r], __int_as_float(bestn[r]));
        }
    }
}

// ---------------------------------------------------------------------------
// Kernel 5: merge per-chunk partials (<=4) and gather the 64-float value row.
// One wave per query row; 8 rows per block.
// ---------------------------------------------------------------------------
__global__ __launch_bounds__(256) void merge_gather_kernel(
    const float*  __restrict__ values,          // n_points x 64
    const float2* __restrict__ partial,         // [n_queries][split]
    float*        __restrict__ out_values_new,  // n_queries x 64
    int split)
{
    const int lane = threadIdx.x & 31;
    const int wave = threadIdx.x >> 5;
    const int row  = blockIdx.x * 8 + wave;

    float d = 3.402823466e38f;
    int   idxn = 0x7fffffff;
    if (lane < split) {
        const float2 p = partial[(size_t)row * split + lane];
        d = p.x;
        idxn = __float_as_int(p.y);
    }
#pragma unroll
    for (int s = 1; s <= 2; s <<= 1) {          // reduce over up to 4 chunks
        const float od = __shfl_xor(d, s, 32);
        const int   on = __shfl_xor(idxn, s, 32);
        const bool take = (od < d) || (od == d && on < idxn);
        d = take ? od : d;
        idxn = take ? on : idxn;
    }
    const int idx = __shfl(idxn, 0, 32);

    const float2* vsrc = (const float2*)(values + (size_t)idx * 64);
    float2*       vdst = (float2*)(out_values_new + (size_t)row * 64);
    vdst[lane] = vsrc[lane];
}

// ---------------------------------------------------------------------------
// Fallback (no workspace): fused NN + gather, B built inline from coords.
// ---------------------------------------------------------------------------
__global__ __launch_bounds__(256) void nn_gather_fused_kernel(
    const float* __restrict__ values,
    const float* __restrict__ coords,
    const float* __restrict__ qcoords,
    const float* __restrict__ shift,
    float* __restrict__ out_values_new,
    int n_points)
{
    const int lane = threadIdx.x & 31;
    const int wave = threadIdx.x >> 5;
    const int col  = lane & 15;
    const int hi   = lane >> 4;
    const int m0   = (blockIdx.x * 8 + wave) * 16;

    const int m  = m0 + col;
    const float qx = qcoords[2 * m + 0] - shift[0];
    const float qy = qcoords[2 * m + 1] - shift[1];
    v2f a;
    a.x = hi ? 1.0f : qx;
    a.y = hi ? 0.0f : qy;

    float bestd[8];
    int   bestn[8];
#pragma unroll
    for (int r = 0; r < 8; ++r) { bestd[r] = 3.402823466e38f; bestn[r] = 0; }

    for (int n0 = 0; n0 < n_points; n0 += 16) {
        const int   n  = n0 + col;
        const float cx = coords[2 * n + 0];
        const float cy = coords[2 * n + 1];
        v2f b;
        b.x = hi ? (cx * cx + cy * cy) : (-2.0f * cx);
        b.y = hi ? 0.0f                : (-2.0f * cy);
        const v8f cz = {0.f, 0.f, 0.f, 0.f, 0.f, 0.f, 0.f, 0.f};
        const v8f d = __builtin_amdgcn_wmma_f32_16x16x4_f32(
            false, a, false, b, (short)0, cz, false, false);
#pragma unroll
        for (int r = 0; r < 8; ++r) {
            const bool lt = d[r] < bestd[r];
            bestn[r] = lt ? n    : bestn[r];
            bestd[r] = lt ? d[r] : bestd[r];
        }
    }
#pragma unroll
    for (int s = 1; s <= 8; s <<= 1) {
#pragma unroll
        for (int r = 0; r < 8; ++r) {
            const float od = __shfl_xor(bestd[r], s, 32);
            const int   on = __shfl_xor(bestn[r], s, 32);
            const bool take = (od < bestd[r]) || (od == bestd[r] && on < bestn[r]);
            bestd[r] = take ? od : bestd[r];
            bestn[r] = take ? on : bestn[r];
        }
    }
#pragma unroll
    for (int rr = 0; rr < 16; ++rr) {
        const int idx = __shfl(bestn[rr & 7], (rr >= 8) ? 16 : 0, 32);
        const float2* vsrc = (const float2*)(values + (size_t)idx * 64);
        float2*       vdst = (float2*)(out_values_new + (size_t)(m0 + rr) * 64);
        vdst[lane] = vsrc[lane];
    }
}

// ---------------------------------------------------------------------------
extern "C" void kernel_launch(void* const* d_in, const int* in_sizes, int n_in,
                              void* d_out, int out_size, void* d_ws, size_t ws_size,
                              hipStream_t stream) {
    const float* values  = (const float*)d_in[0];  // n x 64
    const float* coords  = (const float*)d_in[1];  // n x 2
    const float* spacing = (const float*)d_in[2];  // 2
    const float* shift   = (const float*)d_in[3];  // 2

    const int n = in_sizes[1] / 2;                 // 8192
    const int c = in_sizes[0] / n;                 // 64
    const int nq = 3 * n;                          // 24576 queries
    const int n_tiles = n / 16;                    // 512

    float* out_values = (float*)d_out;                       // 4n x c
    float* out_coords = out_values + (size_t)4 * n * c;      // 4n x 2
    const float* qcoords = out_coords + (size_t)2 * n;       // new_coords
    float* out_values_new = out_values + (size_t)n * c;

    prep_coords_kernel<<<(n + 255) / 256, 256, 0, stream>>>(
        coords, spacing, out_coords, n);

    const int n4 = (n * c) / 4;
    copy_values_kernel<<<(n4 + 255) / 256, 256, 0, stream>>>(
        (const float4*)values, (float4*)out_values, n4);

    // Pick the column-split factor that fits the workspace.
    const size_t bfrag_bytes = (size_t)n_tiles * 32 * sizeof(float2);
    int split = 0;
    for (int s = 4; s >= 1; s >>= 1) {
        if (bfrag_bytes + (size_t)nq * s * sizeof(float2) <= ws_size) { split = s; break; }
    }

    if (split > 0) {
        float2* bfrag   = (float2*)d_ws;
        float2* partial = (float2*)((char*)d_ws + bfrag_bytes);

        bfrag_prep_kernel<<<(n + 255) / 256, 256, 0, stream>>>(coords, bfrag, n);

        const int tasks = (nq / 16) * split;       // wave tasks (divisible by 8)
        nn_partial_kernel<<<tasks / 8, 256, 0, stream>>>(
            (const v2f*)bfrag, qcoords, shift, partial, n_tiles / split, split);

        merge_gather_kernel<<<nq / 8, 256, 0, stream>>>(
            values, partial, out_values_new, split);
    } else {
        nn_gather_fused_kernel<<<nq / 128, 256, 0, stream>>>(
            values, coords, qcoords, shift, out_values_new, n);
    }
}